// Loss_40510131536269
// MI455X (gfx1250) — compile-verified
//
#include <hip/hip_runtime.h>
#include <math.h>

// ---------------- problem constants (fixed by reference setup_inputs) ------
#define TSTEPS   50
#define BB       64      // batch
#define VOCAB    5000
#define TREEDEPTH 7
#define NS       255     // n_states = 2^(D+1)-1
#define NPAD     256
#define NLIST    (TREEDEPTH + 1)   // 8 start/end indices
#define MAXE     1024
#define NTHREADS 512
#define BPB      16      // batch rows per block
#define NBLK     (BB / BPB)
#define EPS_F    1.6038108905486378e-28f   // exp(-64)

typedef __attribute__((ext_vector_type(2))) float v2f;
typedef __attribute__((ext_vector_type(8))) float v8f;

// ---------------- compile-time tree construction (mirrors leaf_interval) ---
struct LRL { int l[NLIST]; int r[NLIST]; int n; };
struct EdgeBuf { int out[MAXE]; int in[MAXE]; int n; };

constexpr int build_tree(int depth, int start_idx, LRL& lr, EdgeBuf& eb) {
  if (depth == 0) { lr.n = 1; lr.l[0] = start_idx; lr.r[0] = start_idx; return start_idx; }
  LRL L{}; LRL R{};
  int last = build_tree(depth - 1, start_idx, L, eb);
  const int my = last + 1;
  last = build_tree(depth - 1, my + 1, R, eb);
  for (int i = 0; i < L.n; ++i)
    for (int j = 0; j < R.n; ++j) { eb.out[eb.n] = L.r[i]; eb.in[eb.n] = R.l[j]; eb.n++; }
  lr.n = L.n + 1;
  lr.l[0] = my; for (int i = 0; i < L.n; ++i) lr.l[i + 1] = L.l[i];
  lr.r[0] = my; for (int i = 0; i < R.n; ++i) lr.r[i + 1] = R.r[i];
  return last;
}

constexpr int count_out() {
  LRL root{}; EdgeBuf eb{};
  build_tree(TREEDEPTH, 0, root, eb);
  bool present[NS] = {};
  for (int e = 0; e < eb.n; ++e) present[eb.out[e]] = true;
  int n = 0;
  for (int v = 0; v < NS; ++v) if (present[v]) ++n;
  return n;
}

constexpr int N_OUT = count_out();            // 247 (all nodes off the global right spine)
constexpr int KPAD  = (N_OUT + 3) & ~3;       // 248, K must be multiple of 4 for wmma
constexpr int KP2   = KPAD / 2;               // 124 packed row-pairs

struct Tables {
  int n_out;
  int start_idxs[NLIST]; int end_idxs[NLIST];
  int out_uniq[NPAD];
  // Incidence matrix (u -> j), transposed + packed as f16 pairs:
  // Bpk[p][j] = f16(A[2p][j]) | f16(A[2p+1][j]) << 16.  Entries are exactly 0/1.
  unsigned int Bpk[KP2][NPAD];
};

constexpr Tables make_tables() {
  Tables t{}; LRL root{}; EdgeBuf eb{};
  build_tree(TREEDEPTH, 0, root, eb);
  for (int i = 0; i < NLIST; ++i) { t.start_idxs[i] = root.l[i]; t.end_idxs[i] = root.r[i]; }
  bool present[NS] = {};
  for (int e = 0; e < eb.n; ++e) present[eb.out[e]] = true;
  int node2u[NS] = {};
  for (int v = 0; v < NS; ++v) node2u[v] = -1;
  int n = 0;
  for (int v = 0; v < NS; ++v) if (present[v]) { t.out_uniq[n] = v; node2u[v] = n; ++n; }
  t.n_out = n;
  for (int e = 0; e < eb.n; ++e) {
    const int u = node2u[eb.out[e]];
    const int j = eb.in[e];
    const unsigned int h = 0x3C00u;           // f16 1.0
    if (u & 1) t.Bpk[u >> 1][j] |= (h << 16);
    else       t.Bpk[u >> 1][j] |= h;
  }
  return t;
}

constexpr Tables c_tbl = make_tables();
__device__ const Tables g_tbl = c_tbl;        // ~128KB constant blob in device .data

// ---------------- kernel 1: gather extracted[t][b][v] = lp[v][b][tgt[t][b]] -
__global__ void ctreec_gather(const float* __restrict__ lp,
                              const int* __restrict__ tgt,
                              float* __restrict__ ext) {
  const int idx = blockIdx.x * blockDim.x + threadIdx.x;
  const int total = TSTEPS * BB * NS;
  if (idx >= total) return;
  const int v  = idx % NS;
  const int r  = idx / NS;
  const int b  = r % BB;
  const int t  = r / BB;
  const int tg = tgt[t * BB + b];
  ext[(size_t)(t * BB + b) * NPAD + v] = lp[((size_t)v * BB + b) * VOCAB + tg];
}

// ---------------- kernel 2: sequential scan, 16 batch rows per block -------
__global__ void __launch_bounds__(NTHREADS, 1)
ctreec_scan(const float* __restrict__ ext,
            const int* __restrict__ lens,
            float* __restrict__ out) {
  __shared__ unsigned int s_B[KP2][NPAD];  // LDS-resident packed incidence (124KB)
  __shared__ float s_prev[BPB][NPAD];      // prev_probs (16 rows)
  __shared__ float s_P[BPB][KPAD];         // log_out, then exp-normalized P
  __shared__ float s_logC[BPB];
  __shared__ float s_acc[BPB];
  __shared__ float s_red[BPB][32];

  const int tid  = threadIdx.x;
  const int b0   = blockIdx.x * BPB;    // first global batch row of this block
  const int wave = tid >> 5;            // 16 waves, one N-tile each
  const int lane = tid & 31;
  const int half = lane >> 4;           // 0: lanes 0-15, 1: lanes 16-31
  const int m    = lane & 15;           // batch row within block / column within tile
  const int col  = wave * 16 + m;       // output state column j

  // one-time: stage packed incidence matrix into LDS (16B vector copies)
  {
    const uint4* src = (const uint4*)&g_tbl.Bpk[0][0];
    uint4* dst = (uint4*)&s_B[0][0];
    constexpr int n16 = KP2 * NPAD / 4;
    for (int i = tid; i < n16; i += NTHREADS) dst[i] = src[i];
  }
  // init state
  for (int i = tid; i < BPB * NPAD; i += NTHREADS) (&s_prev[0][0])[i] = 0.0f;
  for (int i = tid; i < BPB * KPAD; i += NTHREADS) (&s_P[0][0])[i] = 0.0f;
  if (tid < BPB) s_acc[tid] = 0.0f;
  __syncthreads();
  if (tid < BPB) {
    #pragma unroll
    for (int i = 0; i < NLIST; ++i) s_prev[tid][g_tbl.start_idxs[i]] = 1.0f;
  }
  __syncthreads();

  for (int t = 0; t < TSTEPS; ++t) {
    const float* ext_t = ext + (size_t)t * BB * NPAD;

    // ---- phase 1: log_out[b][u] = log_safe(prev[b][out_uniq[u]]) + elp ----
    for (int i = tid; i < BPB * KPAD; i += NTHREADS) {
      const int b = i / KPAD, u = i % KPAD;
      float lc = -3.0e38f;
      if (u < N_OUT) {
        const int node = g_tbl.out_uniq[u];
        const float pv = s_prev[b][node];
        const float ls = (pv < EPS_F) ? -64.0f : logf(pv);
        lc = ls + ext_t[(b0 + b) * NPAD + node];
      }
      s_P[b][u] = lc;
    }
    __syncthreads();

    // ---- phase 2: log_C[b] = logsumexp_u log_out[b][u]  (1 wave per row) --
    {
      const int bl = tid >> 5;
      const int g  = tid & 31;
      float mx = -3.0e38f;
      for (int u = g; u < N_OUT; u += 32) mx = fmaxf(mx, s_P[bl][u]);
      s_red[bl][g] = mx;
      __syncthreads();
      float mm = s_red[bl][0];
      #pragma unroll
      for (int k = 1; k < 32; ++k) mm = fmaxf(mm, s_red[bl][k]);
      float sp = 0.0f;
      for (int u = g; u < N_OUT; u += 32) sp += expf(s_P[bl][u] - mm);
      __syncthreads();
      s_red[bl][g] = sp;
      __syncthreads();
      if (g == 0) {
        float ss = 0.0f;
        #pragma unroll
        for (int k = 0; k < 32; ++k) ss += s_red[bl][k];
        s_logC[bl] = mm + logf(ss);
      }
    }
    __syncthreads();

    // ---- phase 3: P = exp_safe(log_out - log_C); accumulate loss terms ----
    for (int i = tid; i < BPB * KPAD; i += NTHREADS) {
      const int b = i / KPAD, u = i % KPAD;
      float val = 0.0f;
      if (u < N_OUT) {
        const float d = s_P[b][u] - s_logC[b];
        val = (d < -64.0f) ? EPS_F : expf(d);
      }
      s_P[b][u] = val;
    }
    if (tid < BPB) {
      const int b = tid;
      const int len = lens[b0 + b];
      if (t + 1 == len) {                       // end term: logsumexp over end_idxs
        float vals[NLIST]; float mx = -3.0e38f;
        #pragma unroll
        for (int i = 0; i < NLIST; ++i) {
          const int node = g_tbl.end_idxs[i];
          const float pv = s_prev[b][node];
          const float ls = (pv < EPS_F) ? -64.0f : logf(pv);
          vals[i] = ls + ext_t[(b0 + b) * NPAD + node];
          mx = fmaxf(mx, vals[i]);
        }
        float ss = 0.0f;
        #pragma unroll
        for (int i = 0; i < NLIST; ++i) ss += expf(vals[i] - mx);
        s_acc[b] += mx + logf(ss);
      }
      if ((t + 1 < len) && (t < TSTEPS - 1)) s_acc[b] += s_logC[b];
    }
    __syncthreads();

    // ---- phase 4: new_prev = P @ A via V_WMMA_F32_16X16X4_F32 -------------
    // Each wave: one 16(M=batch) x 16(N=state) tile, K = KPAD in steps of 4.
    // All operands LDS-resident: no global latency in the sequential loop.
    {
      v8f c = {};
      for (int k0 = 0; k0 < KPAD; k0 += 4) {
        const int kA = k0 + 2 * half;
        // A fragment (16x4 f32): lane holds P[m][kA], P[m][kA+1]
        v2f a = *(const v2f*)&s_P[m][kA];
        // B fragment (4x16 f32): packed f16 pair (rows kA, kA+1), col = lane%16
        const unsigned int w = s_B[(k0 >> 1) + half][col];
        union { unsigned int u; _Float16 h[2]; } cw; cw.u = w;
        v2f bf; bf.x = (float)cw.h[0]; bf.y = (float)cw.h[1];
        c = __builtin_amdgcn_wmma_f32_16x16x4_f32(
                /*neg_a=*/false, a, /*neg_b=*/false, bf,
                /*c_mod=*/(short)0, c, /*reuse_a=*/false, /*reuse_b=*/false);
      }
      // D layout: VGPR r -> row r (lanes 0-15) / row r+8 (lanes 16-31)
      #pragma unroll
      for (int r = 0; r < 8; ++r) s_prev[r + 8 * half][col] = c[r];
    }
    __syncthreads();
  }

  if (tid < BPB) out[b0 + tid] = -s_acc[tid];
}

// ---------------- launcher -------------------------------------------------
extern "C" void kernel_launch(void* const* d_in, const int* in_sizes, int n_in,
                              void* d_out, int out_size, void* d_ws, size_t ws_size,
                              hipStream_t stream) {
  const float* lp   = (const float*)d_in[0];   // (255, 64, 5000) f32
  const int*   tgt  = (const int*)d_in[1];     // (50, 64) i32
  const int*   lens = (const int*)d_in[2];     // (64,) i32
  float* outp = (float*)d_out;                 // (64,) f32
  float* ext  = (float*)d_ws;                  // (50, 64, 256) f32 staging

  const int total = TSTEPS * BB * NS;
  ctreec_gather<<<(total + 255) / 256, 256, 0, stream>>>(lp, tgt, ext);
  ctreec_scan<<<NBLK, NTHREADS, 0, stream>>>(ext, lens, outp);
}